// DILATEMSEloss_35476429865780
// MI455X (gfx1250) — compile-verified
//
#include <hip/hip_runtime.h>
#include <hip/hip_bf16.h>

typedef __attribute__((ext_vector_type(2))) float v2f;
typedef __attribute__((ext_vector_type(8))) float v8f;

#define SDTW_N 256
#define SDTW_B 64
#define GAMMA_ 0.01f
#define INVG_ 100.0f
#define INF_ 1.0e8f

// LDS layout (floats)
#define OFF_M 0                  // 256x256 D-then-R matrix (in-place transform)
#define OFF_T 65536              // target row (256)
#define OFF_X (65536 + 256)      // input row (256)
#define OFF_E (65536 + 512)      // 3 rotating E-diagonal buffers
#define EB_STRIDE 258
#define SMEM_FLOATS (65536 + 512 + 3 * EB_STRIDE)   // 66822 floats = 267288 B < 320KB/WGP

__global__ void __launch_bounds__(256)
sdtw_fused_kernel(const float* __restrict__ input,
                  const float* __restrict__ target,
                  float* __restrict__ ws) {
  extern __shared__ float smem[];
  float* Msh = smem + OFF_M;
  float* t_s = smem + OFF_T;
  float* x_s = smem + OFF_X;
  float* eb  = smem + OFF_E;

  const int tid = threadIdx.x;
  const int b   = blockIdx.x;

  // stage rows into LDS; zero rotating E buffers
  t_s[tid] = target[b * SDTW_N + tid];
  x_s[tid] = input[b * SDTW_N + tid];
  for (int k = tid; k < 3 * EB_STRIDE; k += 256) eb[k] = 0.f;
  __syncthreads();

  // ---- Phase 1: D tiles via WMMA f32 16x16x4 (rank-1 outer product t * x^T) ----
  // D[m,n] = t[m]^2 + x[n]^2 - 2*t[m]*x[n]; outer product done by the matrix core.
  const int lane = tid & 31;
  const int wave = tid >> 5;
  for (int tIdx = wave; tIdx < 256; tIdx += 8) {
    const int mt = tIdx >> 4, nt = tIdx & 15;
    // A (16x4 f32): only K=0 column nonzero -> lanes 0-15 of VGPR0 hold t-tile
    v2f a;  a.x  = (lane < 16) ? t_s[(mt << 4) + lane] : 0.f;  a.y  = 0.f;
    // B (4x16 f32): only K=0 row nonzero -> lanes 0-15 of VGPR0 hold x-tile
    v2f bm; bm.x = (lane < 16) ? x_s[(nt << 4) + lane] : 0.f;  bm.y = 0.f;
    v8f c = {0.f, 0.f, 0.f, 0.f, 0.f, 0.f, 0.f, 0.f};
    c = __builtin_amdgcn_wmma_f32_16x16x4_f32(false, a, false, bm, (short)0, c,
                                              false, false);
    // C/D layout: vgpr r, lanes 0-15 -> M=r, lanes 16-31 -> M=r+8; N = lane%16
    const int n  = (nt << 4) + (lane & 15);
    const int m0 = (mt << 4) + ((lane & 16) >> 1);
    const float xv = x_s[n];
    const float xn2 = xv * xv;
#pragma unroll
    for (int r = 0; r < 8; ++r) {
      const float tv = t_s[m0 + r];
      Msh[(m0 + r) * SDTW_N + n] = tv * tv + xn2 - 2.f * c[r];
    }
  }
  __syncthreads();

  // ---- Phase 2: forward soft-DTW scan, in-place D -> R ----
  // At diagonal d, predecessors are already R; own cell still holds D.
  const int i = tid + 1;
  for (int d = 2; d <= 2 * SDTW_N; ++d) {
    const int j = d - i;
    if (j >= 1 && j <= SDTW_N) {
      const float Dv = Msh[(i - 1) * SDTW_N + (j - 1)];
      const float rd = (i == 1) ? ((j == 1) ? 0.f : INF_)
                                : ((j == 1) ? INF_ : Msh[(i - 2) * SDTW_N + (j - 2)]);
      const float ru = (i == 1) ? INF_ : Msh[(i - 2) * SDTW_N + (j - 1)];
      const float rl = (j == 1) ? INF_ : Msh[(i - 1) * SDTW_N + (j - 2)];
      const float mn = fminf(rd, fminf(ru, rl));
      const float ssum = __expf((mn - rd) * INVG_) + __expf((mn - ru) * INVG_) +
                         __expf((mn - rl) * INVG_);
      Msh[(i - 1) * SDTW_N + (j - 1)] = Dv + mn - GAMMA_ * __logf(ssum);
    }
    __syncthreads();
  }

  // ---- Phase 3: backward gradient E via 3 rotating diagonals; acc = sum E*(i-j)^2 ----
  float acc = 0.f;
  for (int d = 2 * SDTW_N; d >= 2; --d) {
    float* e0 = eb + (d % 3) * EB_STRIDE;        // diagonal d (write)
    float* e1 = eb + ((d + 1) % 3) * EB_STRIDE;  // diagonal d+1
    float* e2 = eb + ((d + 2) % 3) * EB_STRIDE;  // diagonal d+2
    const int j = d - i;
    if (j >= 1 && j <= SDTW_N) {
      float E;
      if (i == SDTW_N && j == SDTW_N) {
        E = 1.f;
      } else {
        const float Rij = Msh[(i - 1) * SDTW_N + (j - 1)];
        E = 0.f;
        if (i < SDTW_N) {  // successor (i+1, j)
          const float dd = t_s[i] - x_s[j - 1];
          E += e1[i + 1] * __expf((Msh[i * SDTW_N + (j - 1)] - dd * dd - Rij) * INVG_);
        }
        if (j < SDTW_N) {  // successor (i, j+1)
          const float dd = t_s[i - 1] - x_s[j];
          E += e1[i] * __expf((Msh[(i - 1) * SDTW_N + j] - dd * dd - Rij) * INVG_);
        }
        if (i < SDTW_N && j < SDTW_N) {  // successor (i+1, j+1)
          const float dd = t_s[i] - x_s[j];
          E += e2[i + 1] * __expf((Msh[i * SDTW_N + j] - dd * dd - Rij) * INVG_);
        }
      }
      e0[i] = E;
      const float dij = (float)(i - j);
      acc += E * dij * dij;
    }
    __syncthreads();
  }

  // ---- Phase 4: per-batch reductions (temporal sum, mse sum, R[N,N]) ----
  const float dxt  = x_s[tid] - t_s[tid];
  const float msev = dxt * dxt;
  const float Rnn  = Msh[(SDTW_N - 1) * SDTW_N + (SDTW_N - 1)];
  float* red  = eb;        // reuse E buffers as reduction scratch
  float* redm = eb + 256;
  red[tid]  = acc;
  redm[tid] = msev;
  __syncthreads();
  for (int s2 = 128; s2 > 0; s2 >>= 1) {
    if (tid < s2) {
      red[tid]  += red[tid + s2];
      redm[tid] += redm[tid + s2];
    }
    __syncthreads();
  }
  if (tid == 0) {
    ws[b]                = Rnn;      // per-batch soft-DTW value
    ws[SDTW_B + b]       = red[0];   // per-batch sum(E * Omega)
    ws[2 * SDTW_B + b]   = redm[0];  // per-batch sum((x-t)^2)
  }
}

__global__ void __launch_bounds__(64)
sdtw_finalize_kernel(const float* __restrict__ ws, float* __restrict__ out) {
  __shared__ float s0[64], s1[64], s2[64];
  const int t = threadIdx.x;
  s0[t] = ws[t];
  s1[t] = ws[SDTW_B + t];
  s2[t] = ws[2 * SDTW_B + t];
  __syncthreads();
  for (int s = 32; s > 0; s >>= 1) {
    if (t < s) { s0[t] += s0[t + s]; s1[t] += s1[t + s]; s2[t] += s2[t + s]; }
    __syncthreads();
  }
  if (t == 0) {
    const float loss_shape    = s0[0] / (float)SDTW_B;
    const float loss_temporal = s1[0] / ((float)SDTW_B * (float)SDTW_N * (float)SDTW_N);
    const float mse           = s2[0] / ((float)SDTW_B * (float)SDTW_N);
    out[0] = 0.5f * loss_shape + 0.5f * loss_temporal + mse;
  }
}

extern "C" void kernel_launch(void* const* d_in, const int* in_sizes, int n_in,
                              void* d_out, int out_size, void* d_ws, size_t ws_size,
                              hipStream_t stream) {
  (void)in_sizes; (void)n_in; (void)out_size; (void)ws_size;
  const float* input  = (const float*)d_in[0];   // (B, N, 1) float32
  const float* target = (const float*)d_in[1];   // (B, N, 1) float32
  float* ws  = (float*)d_ws;                     // 3*B floats of partials
  float* out = (float*)d_out;                    // scalar

  const size_t smem_bytes = (size_t)SMEM_FLOATS * sizeof(float);  // ~267 KB < 320 KB/WGP
  sdtw_fused_kernel<<<SDTW_B, 256, smem_bytes, stream>>>(input, target, ws);
  sdtw_finalize_kernel<<<1, 64, 0, stream>>>(ws, out);
}